// SparseMixerRouter_65481071411008
// MI455X (gfx1250) — compile-verified
//
#include <hip/hip_runtime.h>
#include <math.h>

#define TOKENS     16384
#define HIDDEN     2048
#define NEXP       64
#define JITTER_EPS 0.01f
#define Z_COEFF    0.001f

// flat output layout (concatenated in reference return order)
#define MULT_OFF  0
#define GATES_OFF (TOKENS * 2)
#define SEL_OFF   (GATES_OFF + TOKENS * NEXP)
#define Z_OFF     (SEL_OFF + TOKENS * 2)

#define TOK_PER_BLOCK 64   // 4 waves x 16 tokens

typedef __attribute__((ext_vector_type(2))) float v2f;
typedef __attribute__((ext_vector_type(8))) float v8f;

__global__ __launch_bounds__(128) void sparsemixer_router_kernel(
    const float* __restrict__ x,   // [TOKENS, HIDDEN]
    const float* __restrict__ w,   // [NEXP,  HIDDEN]
    float* __restrict__ out)
{
    __shared__ float lds[TOK_PER_BLOCK][NEXP + 4];   // logits [tokenLocal][expert]
    __shared__ float s_max1[TOK_PER_BLOCK];
    __shared__ float s_invsum[TOK_PER_BLOCK];

    const int lane    = (int)(threadIdx.x & 31u);
    const int wv      = (int)(threadIdx.x >> 5);       // 0..3 -> token sub-tile
    const int tokBase = (int)blockIdx.x * TOK_PER_BLOCK;
    const int half    = lane >> 4;                     // 0: K={0,1}, 1: K={2,3}
    const int l16     = lane & 15;

    // A fragment (16x4 f32): row = this wave's token, per-lane K pair at 2*half
    const float* aptr = x + (size_t)(tokBase + wv * 16 + l16) * HIDDEN + 2 * half;
    // B fragments: expert row l16 of tile e lives at byte offset e*16*HIDDEN*4
    // (fits in the signed 24-bit instruction offset -> one address VGPR pair)
    const float* bptr = w + (size_t)l16 * HIDDEN + 2 * half;

    v8f acc0 = {}, acc1 = {}, acc2 = {}, acc3 = {};
#pragma unroll 4
    for (int k = 0; k < HIDDEN; k += 4) {
        v2f a  = *(const v2f*)(aptr + k);
        v2f b0 = *(const v2f*)(bptr + k);
        v2f b1 = *(const v2f*)(bptr + 16 * HIDDEN + k);
        v2f b2 = *(const v2f*)(bptr + 32 * HIDDEN + k);
        v2f b3 = *(const v2f*)(bptr + 48 * HIDDEN + k);
        acc0 = __builtin_amdgcn_wmma_f32_16x16x4_f32(false, a, false, b0, (short)0, acc0, false, false);
        acc1 = __builtin_amdgcn_wmma_f32_16x16x4_f32(false, a, false, b1, (short)0, acc1, false, false);
        acc2 = __builtin_amdgcn_wmma_f32_16x16x4_f32(false, a, false, b2, (short)0, acc2, false, false);
        acc3 = __builtin_amdgcn_wmma_f32_16x16x4_f32(false, a, false, b3, (short)0, acc3, false, false);
    }

    // C/D layout: VGPR r <-> M=r (lanes 0-15) or M=r+8 (lanes 16-31), N=lane%16
    {
        const int m = wv * 16 + 8 * half;
#pragma unroll
        for (int r = 0; r < 8; ++r) {
            lds[m + r][ 0 + l16] = acc0[r];
            lds[m + r][16 + l16] = acc1[r];
            lds[m + r][32 + l16] = acc2[r];
            lds[m + r][48 + l16] = acc3[r];
        }
    }
    __syncthreads();

    // ---- routing epilogue: one thread per token (64 of 128 threads) ----
    if (threadIdx.x < TOK_PER_BLOCK) {
        const int t  = (int)threadIdx.x;
        const int gt = tokBase + t;

        // pass 1: top-1 (first-max semantics, like jnp.argmax)
        float max1 = -INFINITY; int s1 = 0;
        for (int e = 0; e < NEXP; ++e) {
            float v = lds[t][e];
            if (v > max1) { max1 = v; s1 = e; }
        }

        // pass 2: full softmax denom + masked denom for top-1 gate
        float sumAll = 0.0f, Z1 = 0.0f;
        for (int e = 0; e < NEXP; ++e) {
            float v  = lds[t][e];
            float ex = expf(v - max1);
            sumAll += ex;
            float factor = fmaxf(fabsf(v), max1);          // scores.abs().clamp(min=mx)
            bool  masked = ((max1 - v) / factor) > (2.0f * JITTER_EPS); // NaN -> false
            if (!masked) Z1 += ex;
        }

        // pass 3: top-2 (exclude s1)
        float mx2 = -INFINITY; int s2 = 0;
        for (int e = 0; e < NEXP; ++e) {
            if (e == s1) continue;
            float v = lds[t][e];
            if (v > mx2) { mx2 = v; s2 = e; }
        }

        // pass 4: masked denom for top-2 gate (factor uses ORIGINAL logits)
        float Z2 = 0.0f;
        for (int e = 0; e < NEXP; ++e) {
            if (e == s1) continue;
            float v = lds[t][e];
            float factor = fmaxf(fabsf(v), mx2);
            bool  masked = ((mx2 - v) / factor) > (2.0f * JITTER_EPS);
            if (!masked) Z2 += expf(v - mx2);
        }

        const float m1 = 1.0f / Z1;   // exp(max1-max1)/Z1
        const float m2 = 1.0f / Z2;   // exp(mx2 - mx2)/Z2

        out[MULT_OFF + (size_t)gt * 2 + 0] = m1;
        out[MULT_OFF + (size_t)gt * 2 + 1] = m2;
        out[SEL_OFF  + (size_t)gt * 2 + 0] = (float)s1;
        out[SEL_OFF  + (size_t)gt * 2 + 1] = (float)s2;

        // z-loss contribution: coeff * lse^2 / TOKENS
        float lse = max1 + logf(sumAll);
        atomicAdd(out + Z_OFF, Z_COEFF * lse * lse * (1.0f / (float)TOKENS));

        s_max1[t]   = max1;
        s_invsum[t] = 1.0f / sumAll;
    }
    __syncthreads();

    // ---- original_gates: all 128 threads, coalesced stores (2 threads/token) ----
    {
        const int tt = (int)threadIdx.x >> 1;          // token 0..63
        const int eg = ((int)threadIdx.x & 1) * 32;    // expert chunk base
        const int gt = tokBase + tt;
        const float mx  = s_max1[tt];
        const float inv = s_invsum[tt];
        float* gout = out + GATES_OFF + (size_t)gt * NEXP;
#pragma unroll
        for (int j = 0; j < 32; ++j) {
            float v = lds[tt][eg + j];
            gout[eg + j] = expf(v - mx) * inv;
        }
    }
}

extern "C" void kernel_launch(void* const* d_in, const int* in_sizes, int n_in,
                              void* d_out, int out_size, void* d_ws, size_t ws_size,
                              hipStream_t stream) {
    (void)in_sizes; (void)n_in; (void)out_size; (void)d_ws; (void)ws_size;
    const float* x = (const float*)d_in[0];
    const float* w = (const float*)d_in[1];
    float* out = (float*)d_out;

    // z-loss accumulator must start at zero every launch (graph-capturable memset)
    hipMemsetAsync(out + Z_OFF, 0, sizeof(float), stream);

    dim3 grid(TOKENS / TOK_PER_BLOCK);
    dim3 block(128);
    hipLaunchKernelGGL(sparsemixer_router_kernel, grid, block, 0, stream, x, w, out);
}